// WaveletReconstruction_28406913696119
// MI455X (gfx1250) — compile-verified
//
#include <hip/hip_runtime.h>
#include <stdint.h>

// Haar (db1) inverse DWT, zero-padding mode.
// Inputs : yl, lh, hl, hh  each [8, 32, 256, 256] f32
// Output : [8, 32, 512, 512] f32
//
// out[2i  ,2j  ] = 0.5*(yl + lh + hl + hh)
// out[2i  ,2j+1] = 0.5*(yl + lh - hl - hh)
// out[2i+1,2j  ] = 0.5*(yl - lh + hl - hh)
// out[2i+1,2j+1] = 0.5*(yl - lh - hl + hh)
//
// Memory-bound (~537 MB total traffic, ~23 us floor at 23.3 TB/s).
// CDNA5 paths: GLOBAL_LOAD_ASYNC_TO_LDS_B128 (th:TH_LOAD_NT) staging,
// s_wait_asynccnt, non-temporal b128 stores (streaming, no L2 retention —
// footprint (537 MB) >> L2 (192 MB) and zero reuse).

#define P_PLANES 256   // B*C = 8*32
#define H_IN     256
#define W_IN     256
#define W_OUT    512
#define TROWS    8     // input rows per block
#define NTHREADS 256

typedef float v4f __attribute__((ext_vector_type(4)));

__global__ __launch_bounds__(NTHREADS) void haar_idwt_kernel(
    const float* __restrict__ yl, const float* __restrict__ lh,
    const float* __restrict__ hl, const float* __restrict__ hh,
    float* __restrict__ out)
{
    // 4 subband tiles of TROWS x W_IN f32 = 32 KB
    __shared__ __align__(16) float smem[4 * TROWS * W_IN];

    const int tid = threadIdx.x;
    const int blk = blockIdx.x;
    const int p   = blk >> 5;        // plane (b*C + c), 0..255
    const int rt  = blk & 31;        // row tile within plane
    const int i0  = rt * TROWS;      // first input row of this tile

    // ---------------- stage: async global -> LDS (b128 per lane, NT) -------
    // 4 subbands x 8 rows x 64 quads = 2048 b128 chunks; 8 per thread.
    #pragma unroll
    for (int t = 0; t < 4; ++t) {
        const float* src = (t == 0) ? yl : (t == 1) ? lh : (t == 2) ? hl : hh;
        #pragma unroll
        for (int half = 0; half < 2; ++half) {
            const int rem  = tid + half * NTHREADS;   // 0..511
            const int row  = rem >> 6;                // 0..7
            const int quad = rem & 63;                // 0..63 (16B chunks)
            const size_t gidx = ((size_t)p * H_IN + (size_t)(i0 + row)) * W_IN
                              + (size_t)quad * 4;
            const float* gaddr = src + gidx;
            const unsigned laddr =
                (unsigned)(uintptr_t)&smem[(t * TROWS + row) * W_IN + quad * 4];
            asm volatile("global_load_async_to_lds_b128 %0, %1, off th:TH_LOAD_NT"
                         :: "v"(laddr), "v"(gaddr)
                         : "memory");
        }
    }
    // Wait for our async copies to land in LDS, then sync the workgroup.
    asm volatile("s_wait_asynccnt 0x0" ::: "memory");
    __syncthreads();

    // ---------------- compute + coalesced non-temporal b128 stores ----------
    // 8 rows x 128 column-pairs = 1024 units; 4 per thread.
    const size_t orow_base = (size_t)p * (2 * H_IN);
    #pragma unroll
    for (int k = 0; k < 4; ++k) {
        const int unit = tid + k * NTHREADS;  // 0..1023
        const int row  = unit >> 7;           // 0..7
        const int pr   = unit & 127;          // column-pair index
        const int j0   = pr * 2;              // even input column

        const float2 a = *(const float2*)&smem[(0 * TROWS + row) * W_IN + j0]; // yl
        const float2 b = *(const float2*)&smem[(1 * TROWS + row) * W_IN + j0]; // lh
        const float2 c = *(const float2*)&smem[(2 * TROWS + row) * W_IN + j0]; // hl
        const float2 d = *(const float2*)&smem[(3 * TROWS + row) * W_IN + j0]; // hh

        v4f r0, r1;
        {   // input column j0 -> output columns 4pr, 4pr+1
            const float e = 0.5f * (a.x + b.x), f = 0.5f * (a.x - b.x);
            const float g = 0.5f * (c.x + d.x), h = 0.5f * (c.x - d.x);
            r0.x = e + g; r0.y = e - g;       // row 2i
            r1.x = f + h; r1.y = f - h;       // row 2i+1
        }
        {   // input column j0+1 -> output columns 4pr+2, 4pr+3
            const float e = 0.5f * (a.y + b.y), f = 0.5f * (a.y - b.y);
            const float g = 0.5f * (c.y + d.y), h = 0.5f * (c.y - d.y);
            r0.z = e + g; r0.w = e - g;
            r1.z = f + h; r1.w = f - h;
        }

        const size_t r0row = orow_base + (size_t)(2 * (i0 + row));
        float* o0 = out + r0row * W_OUT + (size_t)(4 * pr);
        __builtin_nontemporal_store(r0, (v4f*)o0);            // contiguous 512B/wave
        __builtin_nontemporal_store(r1, (v4f*)(o0 + W_OUT));
    }
}

extern "C" void kernel_launch(void* const* d_in, const int* in_sizes, int n_in,
                              void* d_out, int out_size, void* d_ws, size_t ws_size,
                              hipStream_t stream) {
    (void)in_sizes; (void)n_in; (void)out_size; (void)d_ws; (void)ws_size;
    const float* yl = (const float*)d_in[0];
    const float* lh = (const float*)d_in[1];
    const float* hl = (const float*)d_in[2];
    const float* hh = (const float*)d_in[3];
    float* out = (float*)d_out;

    const int blocks = P_PLANES * (H_IN / TROWS);  // 256 * 32 = 8192
    hipLaunchKernelGGL(haar_idwt_kernel, dim3(blocks), dim3(NTHREADS), 0, stream,
                       yl, lh, hl, hh, out);
}